// Time2d_35519379538230
// MI455X (gfx1250) — compile-verified
//
#include <hip/hip_runtime.h>
#include <hip/hip_bf16.h>
#include <stdint.h>

// Problem geometry (fixed by the reference).
#define NXg  512
#define NYg  512
#define NEh  4
#define DIMX 520
#define DIMY 520
#define NPAD (DIMX * DIMY)

// Tile geometry: 256 threads (8 wave32) compute a 64x16 interior tile,
// staging a (64+8)x(16+8) haloed u2 tile in LDS (6.9 KB of 320 KB/WGP).
#define TX 64
#define TY 16
#define HW (TX + 2 * NEh) /* 72 floats; 288 B row stride (16B multiple) */
#define HH (TY + 2 * NEh) /* 24 */
#define ROWV (HW / 4)     /* 18 x b128 groups per row */
#define NV   (HH * ROWV)  /* 432 x b128 transfers per tile */

// CDNA5 async global->LDS path (ASYNCcnt), with a portable fallback so the
// TU always compiles on both toolchains / host pass.
#if defined(__HIP_DEVICE_COMPILE__) && defined(__gfx1250__) && \
    __has_builtin(__builtin_amdgcn_global_load_async_to_lds_b32)
#define USE_ASYNC_LDS 1
#if __has_builtin(__builtin_amdgcn_global_load_async_to_lds_b128)
#define USE_ASYNC_B128 1
#else
#define USE_ASYNC_B128 0
#endif
#else
#define USE_ASYNC_LDS 0
#define USE_ASYNC_B128 0
#endif

#if USE_ASYNC_LDS
typedef int v4i __attribute__((vector_size(16)));
typedef __attribute__((address_space(1))) int g_as_int;  // global, scalar
typedef __attribute__((address_space(3))) int l_as_int;  // LDS, scalar
typedef __attribute__((address_space(1))) v4i g_as_v4i;  // global, int4
typedef __attribute__((address_space(3))) v4i l_as_v4i;  // LDS, int4
#endif

__global__ __launch_bounds__(256) void wave_init_kernel(
    const float* __restrict__ vel, float* __restrict__ scale,
    float* __restrict__ ua, float* __restrict__ ub)
{
    int idx = blockIdx.x * 256 + threadIdx.x;
    if (idx >= NPAD) return;
    int jp = idx / DIMX;          // padded row (y)
    int ip = idx - jp * DIMX;     // padded col (x)
    // vpad = pad(vel.T, 4, edge): vpad[jp][ip] = vel[clamp(ip-4)][clamp(jp-4)]
    int x = ip - NEh; x = x < 0 ? 0 : (x > NXg - 1 ? NXg - 1 : x);
    int y = jp - NEh; y = y < 0 ? 0 : (y > NYg - 1 ? NYg - 1 : y);
    float v = vel[x * NYg + y];
    scale[idx] = v * v * 1.0e-8f;   // (v*DT)^2 / H^2 ; DT=1e-3, H=10
    ua[idx] = 0.0f;
    ub[idx] = 0.0f;
}

__global__ __launch_bounds__(256) void wave_step_kernel(
    const float* __restrict__ u2,   // current field (read)
    float* __restrict__ u13,        // previous field (read) / next field (write, in place)
    const float* __restrict__ scale,
    const float* __restrict__ w,
    const int* __restrict__ sxp, const int* __restrict__ syp,
    const int* __restrict__ ryp,
    float* __restrict__ out, int t)
{
    __shared__ __align__(16) float tile[HH][HW];
    const int tid = threadIdx.x;
    const int ip0 = blockIdx.x * TX;   // padded-grid col of halo origin (64-float aligned)
    const int jp0 = blockIdx.y * TY;   // padded-grid row of halo origin

    // ---- Stage haloed u2 tile into LDS (async on CDNA5) --------------------
    // Row stride DIMX = 520 floats = 2080 B and halo col origin are 16B-aligned,
    // so every 4-float group is 16B-aligned in both global memory and LDS.
    const float* gbase = u2 + jp0 * DIMX + ip0;
#if USE_ASYNC_B128
    for (int i = tid; i < NV; i += 256) {
        int r  = i / ROWV;
        int c  = (i - r * ROWV) * 4;
        const float* g = gbase + r * DIMX + c;
        g_as_v4i* gp = (g_as_v4i*)(unsigned long long)(uintptr_t)g;
        l_as_v4i* lp = (l_as_v4i*)(unsigned)(uintptr_t)&tile[r][c];
        __builtin_amdgcn_global_load_async_to_lds_b128(gp, lp, 0, 0);
    }
#else
    for (int i = tid; i < HW * HH; i += 256) {
        int r = i / HW;
        int c = i - r * HW;
        const float* g = gbase + r * DIMX + c;
#if USE_ASYNC_LDS
        g_as_int* gp = (g_as_int*)(unsigned long long)(uintptr_t)g;
        l_as_int* lp = (l_as_int*)(unsigned)(uintptr_t)&tile[r][c];
        __builtin_amdgcn_global_load_async_to_lds_b32(gp, lp, 0, 0);
#else
        tile[r][c] = *g;
#endif
    }
#endif

    const int lx  = tid & (TX - 1);  // 0..63
    const int lyq = tid >> 6;        // 0..3

    // Overlap: prefetch the u1/scale lines this thread will need.
    {
        const int jp = jp0 + NEh + lyq;
        const int ip = ip0 + NEh + lx;
        __builtin_prefetch(&u13[jp * DIMX + ip], 0, 3);
        __builtin_prefetch(&scale[jp * DIMX + ip], 0, 3);
    }

#if USE_ASYNC_LDS
#if __has_builtin(__builtin_amdgcn_s_wait_asynccnt)
    __builtin_amdgcn_s_wait_asynccnt(0);
#else
    asm volatile("s_wait_asynccnt 0" ::: "memory");
#endif
#endif
    __syncthreads();

    const float wt = w[t];
    const int ix = sxp[0] + NEh;
    const int iy = syp[0] + NEh;
    const int jr = ryp[0] + NEh;

    const float c0 = 2.0f * (-205.0f / 72.0f);
    const float c1 = 8.0f / 5.0f;
    const float c2 = -1.0f / 5.0f;
    const float c3 = 8.0f / 315.0f;
    const float c4 = -1.0f / 560.0f;

#pragma unroll
    for (int r = 0; r < TY / 4; ++r) {
        const int ly = lyq + 4 * r;           // {lyq, lyq+4, lyq+8, lyq+12}
        const int jp = jp0 + NEh + ly;
        const int ip = ip0 + NEh + lx;
        const int lj = ly + NEh;
        const int li = lx + NEh;

        float cen = tile[lj][li];
        float lap = c0 * cen;
        lap += c1 * (tile[lj][li - 1] + tile[lj][li + 1] +
                     tile[lj - 1][li] + tile[lj + 1][li]);
        lap += c2 * (tile[lj][li - 2] + tile[lj][li + 2] +
                     tile[lj - 2][li] + tile[lj + 2][li]);
        lap += c3 * (tile[lj][li - 3] + tile[lj][li + 3] +
                     tile[lj - 3][li] + tile[lj + 3][li]);
        lap += c4 * (tile[lj][li - 4] + tile[lj][li + 4] +
                     tile[lj - 4][li] + tile[lj + 4][li]);

        const int gi = jp * DIMX + ip;
        const float sc = scale[gi];
        float val = 2.0f * cen - u13[gi] + sc * lap;
        // Source injection: src_amp = (vpad[iy][ix]*DT)^2 = scale * H^2.
        if (jp == iy && ip == ix) val += wt * (sc * 100.0f);
        u13[gi] = val;                              // u3 overwrites u1 in place
        if (jp == jr) out[t * NXg + (ip - NEh)] = val;  // record after source add
    }
}

extern "C" void kernel_launch(void* const* d_in, const int* in_sizes, int n_in,
                              void* d_out, int out_size, void* d_ws, size_t ws_size,
                              hipStream_t stream)
{
    const float* vel = (const float*)d_in[0];
    const float* w   = (const float*)d_in[1];
    const int* sxp   = (const int*)d_in[2];
    const int* syp   = (const int*)d_in[3];
    const int* ryp   = (const int*)d_in[4];
    float* out = (float*)d_out;

    float* scale = (float*)d_ws;          // NPAD floats
    float* ua    = scale + NPAD;          // wavefield buffer A (zeroed)
    float* ub    = ua + NPAD;             // wavefield buffer B (zeroed)

    wave_init_kernel<<<(NPAD + 255) / 256, 256, 0, stream>>>(vel, scale, ua, ub);

    const int NT = in_sizes[1];
    dim3 grid(NXg / TX, NYg / TY);        // 8 x 32 = 256 blocks of 8 waves
    for (int t = 0; t < NT; ++t) {
        const float* u2 = (t & 1) ? ub : ua;   // u(t)
        float* u13      = (t & 1) ? ua : ub;   // u(t-1) in, u(t+1) out (in place)
        wave_step_kernel<<<grid, 256, 0, stream>>>(u2, u13, scale, w,
                                                   sxp, syp, ryp, out, t);
    }
}